// IGCNArxiv_59966333387104
// MI455X (gfx1250) — compile-verified
//
#include <hip/hip_runtime.h>

typedef __attribute__((ext_vector_type(2))) float v2f;
typedef __attribute__((ext_vector_type(8))) float v8f;

#define ALPHA_PROP 0.9f
#define EPS_LN 1e-5f
#define HID 256
#define CDIM 40
#define NPART 1024

// ---------------------------------------------------------------------------
// WMMA f32 GEMM: C[M,Nc] = A[M,K] @ B[K,Nc] + bias[Nc]
// One wave32 per block computes a 16 x (16*NTILES) tile with
// V_WMMA_F32_16X16X4_F32, sharing one A fragment across NTILES accumulators.
// Out-of-range rows/cols are CLAMPED for loads (garbage results are computed
// but never stored; WMMA never mixes output cells), so the K loop is
// branch-free.
// Lane mapping per CDNA5 ISA 7.12.2:
//   A 16x4  : lane l -> row (l&15); VGPR0 = K (k + 2*(l>>4)), VGPR1 = K+1
//   B 4x16  : lane l -> col (l&15); VGPR0 = K (k + 2*(l>>4)), VGPR1 = K+1
//   C/D     : VGPR i -> row i + 8*(l>>4), col (l&15)
// ---------------------------------------------------------------------------
template <int KT, int NTILES>
__global__ __launch_bounds__(32)
void gemm_wmma_f32(const float* __restrict__ A, const float* __restrict__ B,
                   const float* __restrict__ bias, float* __restrict__ C,
                   int M, int Nc) {
    const int lane = threadIdx.x;
    const int half = lane >> 4;
    const int l16  = lane & 15;
    const int col0 = blockIdx.x * (16 * NTILES);
    const int row0 = blockIdx.y * 16;

    // Clamped indices for loads (stores use the real indices with guards).
    const int arow = (row0 + l16 < M) ? (row0 + l16) : (M - 1);

    int ncl[NTILES];
    v8f acc[NTILES];
#pragma unroll
    for (int j = 0; j < NTILES; ++j) {
        const int n = col0 + 16 * j + l16;
        ncl[j] = (n < Nc) ? n : (Nc - 1);
        const float bv = bias[ncl[j]];
#pragma unroll
        for (int i = 0; i < 8; ++i) acc[j][i] = bv;
    }

    const float* ap = A + (long)arow * KT + 2 * half;   // advances by 4 per k-step
    const float* bp = B + (long)(2 * half) * Nc;        // advances by 4*Nc per k-step

#pragma unroll 4
    for (int k = 0; k < KT; k += 4) {
        v2f a;
        a.x = ap[0];
        a.y = ap[1];
        ap += 4;
#pragma unroll
        for (int j = 0; j < NTILES; ++j) {
            v2f b;
            b.x = bp[ncl[j]];
            b.y = bp[Nc + ncl[j]];
            acc[j] = __builtin_amdgcn_wmma_f32_16x16x4_f32(
                /*neg_a=*/false, a, /*neg_b=*/false, b,
                /*c_mod=*/(short)0, acc[j], /*reuse_a=*/false, /*reuse_b=*/false);
        }
        bp += 4 * (long)Nc;
    }

#pragma unroll
    for (int i = 0; i < 8; ++i) {
        const int m = row0 + i + 8 * half;
        if (m < M) {
#pragma unroll
            for (int j = 0; j < NTILES; ++j) {
                const int n = col0 + 16 * j + l16;
                if (n < Nc) C[(long)m * Nc + n] = acc[j][i];
            }
        }
    }
}

// ---------------------------------------------------------------------------
// LayerNorm over the node axis (axis 0): per-feature column stats.
// ---------------------------------------------------------------------------
__global__ __launch_bounds__(256)
void zero_f32(float* __restrict__ p, int n) {
    int i = blockIdx.x * blockDim.x + threadIdx.x;
    if (i < n) p[i] = 0.0f;
}

__global__ __launch_bounds__(256)
void col_stats(const float* __restrict__ Z, float* __restrict__ colsum,
               float* __restrict__ colsumsq, int M) {
    const int j = threadIdx.x;  // column 0..255
    float s = 0.f, s2 = 0.f;
    for (int r = blockIdx.x; r < M; r += gridDim.x) {
        const float v = Z[(long)r * HID + j];
        s += v; s2 += v * v;
    }
    atomicAdd(&colsum[j], s);
    atomicAdd(&colsumsq[j], s2);
}

__global__ __launch_bounds__(256)
void ln_finalize(const float* __restrict__ colsum, const float* __restrict__ colsumsq,
                 const float* __restrict__ g, const float* __restrict__ o,
                 float* __restrict__ cola, float* __restrict__ colc, int M) {
    const int j = threadIdx.x;
    const float invM = 1.0f / (float)M;
    const float mu  = colsum[j] * invM;
    float var = colsumsq[j] * invM - mu * mu;
    var = var < 0.f ? 0.f : var;
    const float inv = rsqrtf(var + EPS_LN);
    const float a = g[j] * inv;
    cola[j] = a;
    colc[j] = o[j] - a * mu;
}

__global__ __launch_bounds__(256)
void ln_relu_apply(float* __restrict__ Z, const float* __restrict__ cola,
                   const float* __restrict__ colc, long total) {
    const long i = (long)blockIdx.x * blockDim.x + threadIdx.x;
    if (i < total) {
        const int j = (int)(i & (HID - 1));
        const float v = Z[i] * cola[j] + colc[j];
        Z[i] = v > 0.f ? v : 0.f;
    }
}

// ---------------------------------------------------------------------------
// CG solve of (I - 0.9 * S) x = logits, 40 RHS columns, scalar dots over all.
// scal layout: [0]=rsold [1]=pAp [2]=alpha [3]=rsnew [4]=beta
// ---------------------------------------------------------------------------
__global__ __launch_bounds__(256)
void cg_init(const float* __restrict__ P, float* __restrict__ R,
             float* __restrict__ X, float* __restrict__ partials, long total) {
    __shared__ float sd[256];
    const long stride = (long)gridDim.x * blockDim.x;
    float s = 0.f;
    for (long i = (long)blockIdx.x * blockDim.x + threadIdx.x; i < total; i += stride) {
        const float v = P[i];
        R[i] = v;
        X[i] = 0.f;
        s += v * v;
    }
    sd[threadIdx.x] = s; __syncthreads();
    for (int off = 128; off > 0; off >>= 1) {
        if (threadIdx.x < off) sd[threadIdx.x] += sd[threadIdx.x + off];
        __syncthreads();
    }
    if (threadIdx.x == 0) partials[blockIdx.x] = sd[0];
}

__global__ __launch_bounds__(256)
void edge_scatter(const int* __restrict__ snd, const int* __restrict__ rcv,
                  const float* __restrict__ w, const float* __restrict__ P,
                  float* __restrict__ AP, int E2) {
    const int e = blockIdx.x * blockDim.x + threadIdx.x;
    if (e >= E2) return;
    const float coef = -ALPHA_PROP * w[e];
    // 32-bit row offsets: N*40 < 2^31
    const float4* ps = (const float4*)(P + (unsigned)(snd[e] * CDIM));
    float* ar = AP + (unsigned)(rcv[e] * CDIM);
#pragma unroll
    for (int c = 0; c < CDIM / 4; ++c) {
        const float4 v = ps[c];
        atomicAdd(ar + 4 * c + 0, coef * v.x);
        atomicAdd(ar + 4 * c + 1, coef * v.y);
        atomicAdd(ar + 4 * c + 2, coef * v.z);
        atomicAdd(ar + 4 * c + 3, coef * v.w);
    }
}

__global__ __launch_bounds__(256)
void dot_partial(const float* __restrict__ u, const float* __restrict__ v,
                 float* __restrict__ partials, long total) {
    __shared__ float sd[256];
    const long stride = (long)gridDim.x * blockDim.x;
    float s = 0.f;
    for (long i = (long)blockIdx.x * blockDim.x + threadIdx.x; i < total; i += stride)
        s += u[i] * v[i];
    sd[threadIdx.x] = s; __syncthreads();
    for (int off = 128; off > 0; off >>= 1) {
        if (threadIdx.x < off) sd[threadIdx.x] += sd[threadIdx.x + off];
        __syncthreads();
    }
    if (threadIdx.x == 0) partials[blockIdx.x] = sd[0];
}

// mode 0: scal[0]=sum  (rsold)
// mode 1: scal[1]=sum; scal[2]=alpha=rsold/pAp (guarded)
// mode 2: scal[3]=sum; scal[4]=beta=rsnew/rsold (guarded); scal[0]=rsnew
__global__ __launch_bounds__(256)
void reduce_mode(const float* __restrict__ partials, float* __restrict__ scal, int mode) {
    __shared__ float sd[256];
    float s = 0.f;
    for (int i = threadIdx.x; i < NPART; i += 256) s += partials[i];
    sd[threadIdx.x] = s; __syncthreads();
    for (int off = 128; off > 0; off >>= 1) {
        if (threadIdx.x < off) sd[threadIdx.x] += sd[threadIdx.x + off];
        __syncthreads();
    }
    if (threadIdx.x == 0) {
        const float tot = sd[0];
        if (mode == 0) {
            scal[0] = tot;
        } else if (mode == 1) {
            scal[1] = tot;
            scal[2] = (tot > 1e-30f || tot < -1e-30f) ? scal[0] / tot : 0.f;
        } else {
            scal[3] = tot;
            scal[4] = (scal[0] > 1e-30f) ? tot / scal[0] : 0.f;
            scal[0] = tot;
        }
    }
}

__global__ __launch_bounds__(256)
void cg_update(float* __restrict__ X, float* __restrict__ R,
               const float* __restrict__ P, const float* __restrict__ AP,
               const float* __restrict__ scal, float* __restrict__ partials, long total) {
    __shared__ float sd[256];
    const float alpha = scal[2];
    const long stride = (long)gridDim.x * blockDim.x;
    float s = 0.f;
    for (long i = (long)blockIdx.x * blockDim.x + threadIdx.x; i < total; i += stride) {
        X[i] += alpha * P[i];
        const float rn = R[i] - alpha * AP[i];
        R[i] = rn;
        s += rn * rn;
    }
    sd[threadIdx.x] = s; __syncthreads();
    for (int off = 128; off > 0; off >>= 1) {
        if (threadIdx.x < off) sd[threadIdx.x] += sd[threadIdx.x + off];
        __syncthreads();
    }
    if (threadIdx.x == 0) partials[blockIdx.x] = sd[0];
}

__global__ __launch_bounds__(256)
void cg_pupd(float* __restrict__ P, const float* __restrict__ R,
             const float* __restrict__ scal, long total) {
    const float beta = scal[4];
    const long i = (long)blockIdx.x * blockDim.x + threadIdx.x;
    if (i < total) P[i] = R[i] + beta * P[i];
}

// ---------------------------------------------------------------------------
extern "C" void kernel_launch(void* const* d_in, const int* in_sizes, int n_in,
                              void* d_out, int out_size, void* d_ws, size_t ws_size,
                              hipStream_t stream) {
    const float* Xf = (const float*)d_in[0];
    const int*   snd = (const int*)d_in[1];
    const int*   rcv = (const int*)d_in[2];
    const float* ew  = (const float*)d_in[3];
    const float* W0  = (const float*)d_in[4];
    const float* b0  = (const float*)d_in[5];
    const float* g0  = (const float*)d_in[6];
    const float* o0  = (const float*)d_in[7];
    const float* W1  = (const float*)d_in[8];
    const float* b1  = (const float*)d_in[9];
    const float* g1  = (const float*)d_in[10];
    const float* o1  = (const float*)d_in[11];
    const float* W2  = (const float*)d_in[12];
    const float* b2  = (const float*)d_in[13];

    const int FIN = 128;
    const int N   = in_sizes[0] / FIN;
    const int E2  = in_sizes[1];

    // Workspace layout (floats)
    float* ws   = (float*)d_ws;
    float* H0   = ws;                        // N*256
    float* H1   = ws + (long)N * HID;        // N*256
    float* AUX  = ws + 2L * N * HID;
    float* colsum   = AUX;                   // 256
    float* colsumsq = AUX + HID;             // 256
    float* cola     = AUX + 2 * HID;         // 256
    float* colc     = AUX + 3 * HID;         // 256
    float* partials = AUX + 4 * HID;         // NPART
    float* scal     = partials + NPART;      // 8
    // CG vectors alias the (dead) H0 region: 120 cols used of 256 available.
    float* Pv  = H0;
    float* Rv  = H0 + (long)N * CDIM;
    float* APv = H0 + 2L * N * CDIM;
    float* Xv  = (float*)d_out;

    const long totH = (long)N * HID;
    const long totC = (long)N * CDIM;
    const int  tilesM = (N + 15) / 16;

    // ---- MLP layer 0: H0 = X @ W0 + b0 ; LN(axis0) ; ReLU (in place) ----
    gemm_wmma_f32<128, 4><<<dim3(HID / 64, tilesM), 32, 0, stream>>>(Xf, W0, b0, H0, N, HID);
    zero_f32<<<2, 256, 0, stream>>>(colsum, 2 * HID);
    col_stats<<<1024, 256, 0, stream>>>(H0, colsum, colsumsq, N);
    ln_finalize<<<1, 256, 0, stream>>>(colsum, colsumsq, g0, o0, cola, colc, N);
    ln_relu_apply<<<(int)((totH + 255) / 256), 256, 0, stream>>>(H0, cola, colc, totH);

    // ---- MLP layer 1: H1 = H0 @ W1 + b1 ; LN ; ReLU ----
    gemm_wmma_f32<256, 4><<<dim3(HID / 64, tilesM), 32, 0, stream>>>(H0, W1, b1, H1, N, HID);
    zero_f32<<<2, 256, 0, stream>>>(colsum, 2 * HID);
    col_stats<<<1024, 256, 0, stream>>>(H1, colsum, colsumsq, N);
    ln_finalize<<<1, 256, 0, stream>>>(colsum, colsumsq, g1, o1, cola, colc, N);
    ln_relu_apply<<<(int)((totH + 255) / 256), 256, 0, stream>>>(H1, cola, colc, totH);

    // ---- logits -> Pv (H0 region is dead now) ----
    gemm_wmma_f32<256, 1><<<dim3((CDIM + 15) / 16, tilesM), 32, 0, stream>>>(H1, W2, b2, Pv, N, CDIM);

    // ---- CG: x=0, r=p=logits, rsold=<r,r> ----
    cg_init<<<NPART, 256, 0, stream>>>(Pv, Rv, Xv, partials, totC);
    reduce_mode<<<1, 256, 0, stream>>>(partials, scal, 0);

    const int edgeBlocks = (E2 + 255) / 256;
    const int elemBlocks = (int)((totC + 255) / 256);
    for (int it = 0; it < 100; ++it) {
        // Ap = p, then Ap -= 0.9 * S p via atomic scatter
        hipMemcpyAsync(APv, Pv, totC * sizeof(float), hipMemcpyDeviceToDevice, stream);
        edge_scatter<<<edgeBlocks, 256, 0, stream>>>(snd, rcv, ew, Pv, APv, E2);
        // pAp, alpha
        dot_partial<<<NPART, 256, 0, stream>>>(Pv, APv, partials, totC);
        reduce_mode<<<1, 256, 0, stream>>>(partials, scal, 1);
        // x += alpha p ; r -= alpha Ap ; rsnew partials
        cg_update<<<NPART, 256, 0, stream>>>(Xv, Rv, Pv, APv, scal, partials, totC);
        reduce_mode<<<1, 256, 0, stream>>>(partials, scal, 2);
        // p = r + beta p
        cg_pupd<<<elemBlocks, 256, 0, stream>>>(Pv, Rv, scal, totC);
    }
}